// HMM_25529285607738
// MI455X (gfx1250) — compile-verified
//
#include <hip/hip_runtime.h>
#include <math.h>

#define Nn 64
#define Tt 256
#define Ee 128
#define Hh 64
#define Vv 10000
#define STEPS 255 /* T-1 */

typedef __attribute__((ext_vector_type(2))) float v2f;
typedef __attribute__((ext_vector_type(8))) float v8f;

// ---------------------------------------------------------------------------
// Phase 1: column log-softmax stats of emit_w over V (axis=0).
// colmax[h], collse[h] = log sum_v exp(emit_w[v,h] - colmax[h])
// ---------------------------------------------------------------------------
__global__ __launch_bounds__(1024)
void emit_lse_kernel(const float* __restrict__ emit_w,
                     float* __restrict__ colmax,
                     float* __restrict__ collse) {
  __shared__ float red[16][64];
  const int h = threadIdx.x & 63;
  const int g = threadIdx.x >> 6; // 0..15
  float m = -INFINITY;
  for (int v = g; v < Vv; v += 16) m = fmaxf(m, emit_w[v * Hh + h]);
  red[g][h] = m;
  __syncthreads();
  if (g == 0) {
    float mm = red[0][h];
#pragma unroll
    for (int i = 1; i < 16; ++i) mm = fmaxf(mm, red[i][h]);
    red[0][h] = mm;
    colmax[h] = mm;
  }
  __syncthreads();
  const float cm = red[0][h];
  __syncthreads(); // everyone has read cm before red is reused
  float s = 0.f;
  for (int v = g; v < Vv; v += 16) s += __expf(emit_w[v * Hh + h] - cm);
  red[g][h] = s;
  __syncthreads();
  if (g == 0) {
    float ss = red[0][h];
#pragma unroll
    for (int i = 1; i < 16; ++i) ss += red[i][h];
    collse[h] = __logf(ss);
  }
}

// ---------------------------------------------------------------------------
// Phase 2: WMMA GEMM + fused row-group log-softmax.
// Rows = (s_local, n) pairs (chunk of timesteps), cols = (i,j) flat = 4096.
// Each block: 32 rows x 64 cols (exactly one softmax group i per row).
// 8 waves, each one V_WMMA_F32_16X16X4_F32 tile, K=128 in steps of 4.
// ---------------------------------------------------------------------------
__global__ __launch_bounds__(256)
void trans_kernel(const int* __restrict__ words,
                  const float* __restrict__ embed_w,
                  const float* __restrict__ trans_w,
                  const float* __restrict__ trans_b,
                  float* __restrict__ LT, // [csz*64][4096]
                  int s0) {
  __shared__ __align__(16) float As[32][132];
  __shared__ __align__(16) float Bs[64][132];
  __shared__ float Ds[32][65];
  __shared__ float red[32][8];
  __shared__ int wrow[32];

  const int tid = threadIdx.x;
  const int col_base = blockIdx.x * 64;
  const int row_base = blockIdx.y * 32;

  if (tid < 32) {
    const int rg = row_base + tid;
    const int sl = rg >> 6, n = rg & 63;
    wrow[tid] = words[n * Tt + (s0 + sl)]; // input embedding word at step s
  }
  __syncthreads();

  // Load A tile (gathered embeddings): 32 rows x 128
  {
    const int r = tid >> 3;  // 0..31
    const int kq = tid & 7;  // 8 threads per row, 16 floats each
    const float* src = embed_w + (size_t)wrow[r] * Ee;
#pragma unroll
    for (int q = 0; q < 4; ++q) {
      const int k = kq * 16 + q * 4;
      *(float4*)&As[r][k] = *(const float4*)(src + k);
    }
  }
  // Load B tile (trans_w rows = output cols): 64 x 128
  {
    const int c = tid >> 2;  // 0..63
    const int kq = tid & 3;  // 4 threads per row, 32 floats each
    const float* src = trans_w + (size_t)(col_base + c) * Ee;
#pragma unroll
    for (int q = 0; q < 8; ++q) {
      const int k = kq * 32 + q * 4;
      *(float4*)&Bs[c][k] = *(const float4*)(src + k);
    }
  }
  __syncthreads();

  // WMMA: wave -> one 16x16 tile. 2x4 wave layout over 32x64 block tile.
  const int wave = tid >> 5;
  const int lane = tid & 31;
  const int mt = wave >> 2; // 0..1
  const int nt = wave & 3;  // 0..3
  const int kh = lane >> 4; // K-half select per 16x16x4 f32 layout
  const int ml = lane & 15;

  v8f acc = {0.f, 0.f, 0.f, 0.f, 0.f, 0.f, 0.f, 0.f};
  const float* arow = &As[16 * mt + ml][0];
  const float* brow = &Bs[16 * nt + ml][0];
#pragma unroll
  for (int kk = 0; kk < Ee; kk += 4) {
    v2f a, b;
    a.x = arow[kk + 2 * kh];
    a.y = arow[kk + 2 * kh + 1];
    b.x = brow[kk + 2 * kh];
    b.y = brow[kk + 2 * kh + 1];
    acc = __builtin_amdgcn_wmma_f32_16x16x4_f32(
        /*neg_a=*/false, a, /*neg_b=*/false, b,
        /*c_mod=*/(short)0, acc, /*reuse_a=*/false, /*reuse_b=*/false);
  }

  // Spill D to LDS per the 16x16 f32 C/D layout (VGPR r: M=r / M=8+r).
#pragma unroll
  for (int r = 0; r < 8; ++r)
    Ds[16 * mt + 8 * kh + r][16 * nt + ml] = acc[r];
  __syncthreads();

  // Fused bias + log-softmax over the 64-column group, per row.
  const int r = tid >> 3;  // 0..31
  const int seg = tid & 7; // 8 cols per thread
  float vals[8];
#pragma unroll
  for (int i = 0; i < 8; ++i) {
    const int c = seg * 8 + i;
    vals[i] = Ds[r][c] + trans_b[col_base + c];
  }
  float m = vals[0];
#pragma unroll
  for (int i = 1; i < 8; ++i) m = fmaxf(m, vals[i]);
  red[r][seg] = m;
  __syncthreads();
  if (seg == 0) {
    float mm = red[r][0];
#pragma unroll
    for (int i = 1; i < 8; ++i) mm = fmaxf(mm, red[r][i]);
    red[r][0] = mm;
  }
  __syncthreads();
  const float rowm = red[r][0];
  float s = 0.f;
#pragma unroll
  for (int i = 0; i < 8; ++i) s += __expf(vals[i] - rowm);
  __syncthreads(); // all read rowm before red reuse
  red[r][seg] = s;
  __syncthreads();
  if (seg == 0) {
    float ss = red[r][0];
#pragma unroll
    for (int i = 1; i < 8; ++i) ss += red[r][i];
    red[r][0] = rowm + __logf(ss);
  }
  __syncthreads();
  const float lse = red[r][0];
  float* out = LT + (size_t)(row_base + r) * 4096 + col_base + seg * 8;
#pragma unroll
  for (int i = 0; i < 8; ++i) out[i] = vals[i] - lse;
}

// ---------------------------------------------------------------------------
// Phase 3: sequential scan. One block per sequence n; 256 threads:
// j = tid&63 (target state), g = tid>>6 (4-way split of the i-contraction).
// ---------------------------------------------------------------------------
__global__ __launch_bounds__(256)
void scan_kernel(const int* __restrict__ words,
                 const float* __restrict__ hidden_state,
                 const float* __restrict__ emit_w,
                 const float* __restrict__ colmax,
                 const float* __restrict__ collse,
                 const float* __restrict__ LT,
                 float* __restrict__ acc_state,
                 float* __restrict__ logh_state,
                 float* __restrict__ d_out,
                 int s0, int csz) {
  __shared__ float lh[64];
  __shared__ float pm[4][64];
  __shared__ float ps[4][64];
  __shared__ float jbuf[64];
  __shared__ float r8[8];
  __shared__ float ellS;

  const int n = blockIdx.x;
  const int tid = threadIdx.x;
  const int j = tid & 63;
  const int g = tid >> 6; // 0..3

  if (tid < 64)
    lh[j] = (s0 == 0) ? __logf(hidden_state[n * Hh + j]) : logh_state[n * Hh + j];
  float acc = (s0 == 0) ? 0.f : acc_state[n];
  __syncthreads();

  const float ecm = colmax[j];
  const float ecl = collse[j];

  for (int sl = 0; sl < csz; ++sl) {
    const int s = s0 + sl;
    const float* Lrow = LT + ((size_t)sl * 64 + n) * 4096;

    // partial online-lse over i in [16g, 16g+16)
    float m = -INFINITY, sum = 0.f;
#pragma unroll
    for (int q = 0; q < 16; ++q) {
      const int i = g * 16 + q;
      const float x = lh[i] + Lrow[i * 64 + j];
      const float nm = fmaxf(m, x);
      sum = sum * __expf(m - nm) + __expf(x - nm);
      m = nm;
    }
    pm[g][j] = m;
    ps[g][j] = sum;
    __syncthreads();

    if (g == 0) {
      const float M = fmaxf(fmaxf(pm[0][j], pm[1][j]), fmaxf(pm[2][j], pm[3][j]));
      const float S = ps[0][j] * __expf(pm[0][j] - M) + ps[1][j] * __expf(pm[1][j] - M) +
                      ps[2][j] * __expf(pm[2][j] - M) + ps[3][j] * __expf(pm[3][j] - M);
      const float hid = M + __logf(S);
      const int w = words[n * Tt + s + 1]; // emitted word
      jbuf[j] = hid + emit_w[(size_t)w * Hh + j] - ecm - ecl;
    }
    __syncthreads();

    // lse over jbuf[64]: 8-thread partials, then scalar combine
    if (tid < 8) {
      float mm = jbuf[tid * 8];
#pragma unroll
      for (int q = 1; q < 8; ++q) mm = fmaxf(mm, jbuf[tid * 8 + q]);
      r8[tid] = mm;
    }
    __syncthreads();
    if (tid == 0) {
      float mm = r8[0];
#pragma unroll
      for (int q = 1; q < 8; ++q) mm = fmaxf(mm, r8[q]);
      r8[0] = mm;
    }
    __syncthreads();
    const float jm = r8[0];
    __syncthreads(); // all read jm before r8 reuse
    if (tid < 8) {
      float ss = 0.f;
#pragma unroll
      for (int q = 0; q < 8; ++q) ss += __expf(jbuf[tid * 8 + q] - jm);
      r8[tid] = ss;
    }
    __syncthreads();
    if (tid == 0) {
      float ss = r8[0];
#pragma unroll
      for (int q = 1; q < 8; ++q) ss += r8[q];
      ellS = jm + __logf(ss);
    }
    __syncthreads();
    const float ell = ellS;
    acc += ell;
    if (tid < 64) lh[j] = jbuf[j] - ell;
    __syncthreads();
  }

  if (tid < 64) logh_state[n * Hh + j] = lh[j];
  if (tid == 0) acc_state[n] = acc;
  if (s0 + csz == STEPS) {
    if (tid == 0) d_out[n] = acc;                 // emit_marginal
    if (tid < 64) d_out[64 + n * 64 + j] = lh[j]; // final log_h
  }
}

// ---------------------------------------------------------------------------
extern "C" void kernel_launch(void* const* d_in, const int* in_sizes, int n_in,
                              void* d_out, int out_size, void* d_ws, size_t ws_size,
                              hipStream_t stream) {
  (void)in_sizes; (void)n_in; (void)out_size;
  const int* words = (const int*)d_in[0];
  const float* hidden = (const float*)d_in[1];
  const float* embed_w = (const float*)d_in[2];
  const float* trans_w = (const float*)d_in[3];
  const float* trans_b = (const float*)d_in[4];
  const float* emit_w = (const float*)d_in[5];
  float* out = (float*)d_out;

  float* ws = (float*)d_ws;
  float* colmax = ws;            // 64
  float* collse = ws + 64;       // 64
  float* acc_state = ws + 128;   // 64
  float* logh_state = ws + 192;  // 4096
  float* LT = ws + 8192;         // chunk buffer: C * 64 * 4096 floats

  const size_t perStepBytes = (size_t)64 * 4096 * sizeof(float); // 1 MiB/step
  const size_t hdrBytes = (size_t)8192 * sizeof(float);
  const size_t avail = ws_size > hdrBytes ? ws_size - hdrBytes : 0;
  int C = (int)(avail / perStepBytes);
  if (C < 1) C = 1;
  if (C > STEPS) C = STEPS;

  emit_lse_kernel<<<1, 1024, 0, stream>>>(emit_w, colmax, collse);

  for (int s0 = 0; s0 < STEPS; s0 += C) {
    const int csz = (STEPS - s0 < C) ? (STEPS - s0) : C;
    dim3 grid(4096 / 64, csz * 2); // 64 col-blocks x (csz*64/32) row-blocks
    trans_kernel<<<grid, 256, 0, stream>>>(words, embed_w, trans_w, trans_b, LT, s0);
    scan_kernel<<<64, 256, 0, stream>>>(words, hidden, emit_w, colmax, collse,
                                        LT, acc_state, logh_state, out, s0, csz);
  }
}